// NeuralTheoremProver_10462540333431
// MI455X (gfx1250) — compile-verified
//
#include <hip/hip_runtime.h>

#define E_DIM 8192
#define B_DIM 1024
#define R_DIM 16
#define D_DIM 128
#define ZSPLIT 16          // gridDim.y
#define TPB 256            // 8 waves per block
#define WPB (TPB / 32)
#define LDS_STRIDE 132     // 128 + 4-dword pad -> conflict-free WMMA-A reads

typedef __attribute__((ext_vector_type(2))) float v2f;
typedef __attribute__((ext_vector_type(8))) float v8f;

__device__ __forceinline__ float dot4(float4 a, float4 b) {
    return a.x * b.x + a.y * b.y + a.z * b.z + a.w * b.w;
}

// full-wave (32 lane) xor-tree reduction; result valid in all lanes
__device__ __forceinline__ float wred(float v) {
    v += __shfl_xor(v, 16, 32);
    v += __shfl_xor(v, 8, 32);
    v += __shfl_xor(v, 4, 32);
    v += __shfl_xor(v, 2, 32);
    v += __shfl_xor(v, 1, 32);
    return v;
}

// ---------------------------------------------------------------------------
// Per-z precompute: Zr[z,r] = zz[z] - 2*rz[r,z],  Z2[z] = zz[z] + 2*zq[z]
// One wave per z row; lane holds float4 of the 128-dim row.
// ---------------------------------------------------------------------------
__global__ __launch_bounds__(128) void precompute_z(
    const float* __restrict__ Etab, const float* __restrict__ Rules,
    const int* __restrict__ qrel_p, float* __restrict__ Zr, float* __restrict__ Z2) {
    const int wave = threadIdx.x >> 5, lane = threadIdx.x & 31;
    const int z = blockIdx.x * 4 + wave;
    float4 ev = ((const float4*)(Etab + (size_t)z * D_DIM))[lane];
    float zz = wred(dot4(ev, ev));
    const int qrel = *qrel_p;
    float4 qv = ((const float4*)(Rules + (size_t)qrel * D_DIM))[lane];
    float zq = wred(dot4(ev, qv));
    if (lane == 0) Z2[z] = zz + 2.0f * zq;
#pragma unroll
    for (int r = 0; r < R_DIM; r++) {
        float4 rv = ((const float4*)(Rules + (size_t)r * D_DIM))[lane];
        float rz = wred(dot4(ev, rv));
        if (lane == 0) Zr[z * R_DIM + r] = zz - 2.0f * rz;
    }
}

// ---------------------------------------------------------------------------
// Per-b precompute: A1[b,r] = hh+rr[r]+2*hr[b,r],  T2[b] = qq+tt[b]-2*qt[b]
// One wave per batch element b.
// ---------------------------------------------------------------------------
__global__ __launch_bounds__(128) void precompute_b(
    const float* __restrict__ Etab, const float* __restrict__ Rules,
    const int* __restrict__ head, const int* __restrict__ tail,
    const int* __restrict__ qrel_p, float* __restrict__ A1, float* __restrict__ T2) {
    const int wave = threadIdx.x >> 5, lane = threadIdx.x & 31;
    const int b = blockIdx.x * 4 + wave;
    const int hidx = head[b], tidx = tail[b];
    float4 hv = ((const float4*)(Etab + (size_t)hidx * D_DIM))[lane];
    float4 tv = ((const float4*)(Etab + (size_t)tidx * D_DIM))[lane];
    float hh = wred(dot4(hv, hv));
    float tt = wred(dot4(tv, tv));
    const int qrel = *qrel_p;
    float4 qv = ((const float4*)(Rules + (size_t)qrel * D_DIM))[lane];
    float qt = wred(dot4(tv, qv));
    float qq = wred(dot4(qv, qv));
    if (lane == 0) T2[b] = qq + tt - 2.0f * qt;
#pragma unroll
    for (int r = 0; r < R_DIM; r++) {
        float4 rv = ((const float4*)(Rules + (size_t)r * D_DIM))[lane];
        float hr = wred(dot4(hv, rv));
        float rr = wred(dot4(rv, rv));
        if (lane == 0) A1[b * R_DIM + r] = hh + rr + 2.0f * hr;
    }
}

// ---------------------------------------------------------------------------
// Main fused kernel. Block = 8 waves, one 16-row b-tile shared via LDS.
// Per wave: 16(b) x 16(z) tiles; hz/zt via V_WMMA_F32_16X16X4_F32 with the
// A operands (h/t tiles) read from padded LDS (bank-conflict-free b64 reads)
// and the B operand (entity rows) streamed from L2.
// ---------------------------------------------------------------------------
__global__ __launch_bounds__(TPB) void ntp_main(
    const float* __restrict__ Etab, const int* __restrict__ head,
    const int* __restrict__ tail, const float* __restrict__ Zr,
    const float* __restrict__ Z2, const float* __restrict__ A1,
    const float* __restrict__ T2, float* __restrict__ out) {
    __shared__ float sH[16 * LDS_STRIDE];
    __shared__ float sT[16 * LDS_STRIDE];
    __shared__ float sA1[16 * 16];
    __shared__ float sT2[16];

    const int bt = blockIdx.x * 16;
    const int tid = threadIdx.x;

    // Stage A-operand tiles (block-uniform) + per-row constants into LDS.
    for (int idx = tid; idx < 16 * 32; idx += TPB) {
        const int row = idx >> 5, c4 = idx & 31;
        float4 hv = ((const float4*)(Etab + (size_t)head[bt + row] * D_DIM))[c4];
        float4 tv = ((const float4*)(Etab + (size_t)tail[bt + row] * D_DIM))[c4];
        float* hp = &sH[row * LDS_STRIDE + c4 * 4];
        float* tp = &sT[row * LDS_STRIDE + c4 * 4];
        hp[0] = hv.x; hp[1] = hv.y; hp[2] = hv.z; hp[3] = hv.w;
        tp[0] = tv.x; tp[1] = tv.y; tp[2] = tv.z; tp[3] = tv.w;
    }
    if (tid < 256) sA1[tid] = A1[bt * R_DIM + tid];
    if (tid < 16) sT2[tid] = T2[bt + tid];
    __syncthreads();

    const int wave = tid >> 5, lane = tid & 31;
    const int half = lane >> 4, ln = lane & 15;

    // WMMA A-operand base (16x4 f32 layout): lane<16 -> K0,K1; lane>=16 -> K2,K3
    const float* hlds = &sH[ln * LDS_STRIDE + 2 * half];
    const float* tlds = &sT[ln * LDS_STRIDE + 2 * half];

    float accum[8];
#pragma unroll
    for (int i = 0; i < 8; i++) accum[i] = 0.0f;

    const int zbase = blockIdx.y * (E_DIM / ZSPLIT) + wave * (E_DIM / ZSPLIT / WPB);
    const int ntiles = E_DIM / ZSPLIT / WPB / 16;  // 4

    for (int zt_i = 0; zt_i < ntiles; zt_i++) {
        const int z0 = zbase + zt_i * 16;
        // B-operand (4x16 f32 layout): lane holds column N=ln = entity row z0+ln
        const float* ep = Etab + (size_t)(z0 + ln) * D_DIM + 2 * half;
        v8f chz = {};
        v8f czt = {};
#pragma unroll
        for (int k = 0; k < 32; k++) {
            v2f bv = *(const v2f*)(ep + 4 * k);
            v2f ah = *(const v2f*)(hlds + 4 * k);
            v2f at = *(const v2f*)(tlds + 4 * k);
            chz = __builtin_amdgcn_wmma_f32_16x16x4_f32(false, ah, false, bv,
                                                        (short)0, chz, false, false);
            czt = __builtin_amdgcn_wmma_f32_16x16x4_f32(false, at, false, bv,
                                                        (short)0, czt, false, false);
        }

        // Epilogue: C layout — VGPR i holds M=i (lanes 0-15) / M=8+i (lanes 16-31), N=ln
        const int myz = z0 + ln;
        float zr[16];
        const float4* zrp = (const float4*)(Zr + (size_t)myz * R_DIM);
#pragma unroll
        for (int j = 0; j < 4; j++) {
            float4 v = zrp[j];
            zr[4 * j + 0] = v.x; zr[4 * j + 1] = v.y;
            zr[4 * j + 2] = v.z; zr[4 * j + 3] = v.w;
        }
        const float z2v = Z2[myz];
#pragma unroll
        for (int i = 0; i < 8; i++) {
            const int m = i + 8 * half;
            const float hz = chz[i];
            const float ztv = czt[i];
            // sq2 = Z2[z] + T2[b] - 2*zt ;  s2 = -sqrt(max(sq2,0))
            float sq2 = z2v + sT2[m] - 2.0f * ztv;
            float s2m = sqrtf(fmaxf(sq2, 0.0f));  // |s2|
            float ssum = 0.0f;                    // |S1| = sum_r sqrt(...)
#pragma unroll
            for (int r = 0; r < R_DIM; r++) {
                float t = (sA1[m * 16 + r] + zr[r]) - 2.0f * hz;
                ssum += sqrtf(fmaxf(t, 0.0f));
            }
            accum[i] += ssum * s2m;  // (-|S1|)*(-|s2|) = |S1|*|s2|
        }
    }

    // Reduce across the 16 N-lanes of each half-wave, then atomic per row
#pragma unroll
    for (int i = 0; i < 8; i++) {
        float v = accum[i];
        v += __shfl_xor(v, 8, 32);
        v += __shfl_xor(v, 4, 32);
        v += __shfl_xor(v, 2, 32);
        v += __shfl_xor(v, 1, 32);
        if (ln == 0) atomicAdd(&out[bt + i + 8 * half], v * (1.0f / (float)E_DIM));
    }
}

extern "C" void kernel_launch(void* const* d_in, const int* in_sizes, int n_in,
                              void* d_out, int out_size, void* d_ws, size_t ws_size,
                              hipStream_t stream) {
    const int* head = (const int*)d_in[0];
    const int* tail = (const int*)d_in[1];
    const int* qrel = (const int*)d_in[2];
    // d_in[3] = depth (unused at depth==1)
    const float* Etab = (const float*)d_in[4];
    const float* Rules = (const float*)d_in[5];

    float* ws = (float*)d_ws;
    float* Zr = ws;                      // E * R
    float* Z2 = Zr + E_DIM * R_DIM;      // E
    float* A1 = Z2 + E_DIM;              // B * R
    float* T2 = A1 + B_DIM * R_DIM;      // B
    float* out = (float*)d_out;

    hipMemsetAsync(out, 0, B_DIM * sizeof(float), stream);
    precompute_z<<<E_DIM / 4, 128, 0, stream>>>(Etab, Rules, qrel, Zr, Z2);
    precompute_b<<<B_DIM / 4, 128, 0, stream>>>(Etab, Rules, head, tail, qrel, A1, T2);
    ntp_main<<<dim3(B_DIM / 16, ZSPLIT), TPB, 0, stream>>>(Etab, head, tail, Zr, Z2,
                                                           A1, T2, out);
}